// PyramidGraphSage_72541997629777
// MI455X (gfx1250) — compile-verified
//
#include <hip/hip_runtime.h>
#include <math.h>

// PyramidGraphSage on MI455X (gfx1250, wave32).
// Memory-bound on adj (134MB, fits in 192MB L2) -> fp32 everywhere,
// matrix math via V_WMMA_F32_16X16X4_F32, adj/feature tiles staged with
// GLOBAL_LOAD_ASYNC_TO_LDS_B128 + s_wait_asynccnt double buffering.

#define NN 4096
#define FF 128
#define BB 2

typedef __attribute__((ext_vector_type(2))) float v2f;
typedef __attribute__((ext_vector_type(8))) float v8f;

__device__ __forceinline__ v8f wmma4(v2f a, v2f b, v8f c) {
  // 8 args: (neg_a, A, neg_b, B, c_mod, C, reuse_a, reuse_b)
  return __builtin_amdgcn_wmma_f32_16x16x4_f32(false, a, false, b, (short)0, c,
                                               false, false);
}

// generic->LDS(as3) cast: yields the 32-bit LDS byte offset the async
// instructions want in their VDST operand.
__device__ __forceinline__ unsigned lds_addr(const void* p) {
  return (unsigned)(unsigned long long)(uintptr_t)(
      __attribute__((address_space(3))) const void*)p;
}

// async 16B copy global -> LDS (per lane), tracked by ASYNCcnt
__device__ __forceinline__ void async_cp16(const float* lds_dst,
                                           const float* gsrc) {
  unsigned loff = lds_addr(lds_dst);
  unsigned long long g = (unsigned long long)(uintptr_t)gsrc;
  asm volatile("global_load_async_to_lds_b128 %0, %1, off"
               :
               : "v"(loff), "v"(g)
               : "memory");
}

__device__ __forceinline__ void wait_async0() {
#if __has_builtin(__builtin_amdgcn_s_wait_asynccnt)
  __builtin_amdgcn_s_wait_asynccnt(0);
#else
  asm volatile("s_wait_asynccnt 0x0" ::: "memory");
#endif
}
__device__ __forceinline__ void wait_async5() {
#if __has_builtin(__builtin_amdgcn_s_wait_asynccnt)
  __builtin_amdgcn_s_wait_asynccnt(5);
#else
  asm volatile("s_wait_asynccnt 0x5" ::: "memory");
#endif
}

// ---------------- norms ----------------
__global__ void rownorm_kernel(const float* __restrict__ adj,
                               float* __restrict__ invr) {
  const int w = threadIdx.x >> 5, lane = threadIdx.x & 31;
  const int b = blockIdx.y;
  const int row = blockIdx.x * 8 + w;
  const float* ap = adj + ((long)b * NN + row) * NN;
  float s = 0.f;
  for (int p = lane; p < NN; p += 32) { float v = ap[p]; s = fmaf(v, v, s); }
  for (int m = 16; m; m >>= 1) s += __shfl_xor(s, m, 32);
  if (lane == 0) invr[b * NN + row] = 1.f / fmaxf(sqrtf(s), 1e-12f);
}

__global__ void colnorm_kernel(const float* __restrict__ adj,
                               float* __restrict__ invc) {
  const int b = blockIdx.y;
  const int j = blockIdx.x * 256 + threadIdx.x;
  const float* ap = adj + (long)b * NN * NN + j;
  float s = 0.f;
  for (int i = 0; i < NN; ++i) { float v = ap[(long)i * NN]; s = fmaf(v, v, s); }
  invc[b * NN + j] = 1.f / fmaxf(sqrtf(s), 1e-12f);
}

// ---------------- segment-gather GEMM ----------------
// out[M,128] = concat(s0,s1,s2)[M, nseg*128] @ W[nseg*128,128] + bias (opt tanh)
__global__ void gemm_seg_kernel(const float* __restrict__ s0,
                                const float* __restrict__ s1,
                                const float* __restrict__ s2,
                                const float* __restrict__ W,
                                const float* __restrict__ bias,
                                float* __restrict__ out,
                                int nseg, int act_tanh) {
  const int lane = threadIdx.x & 31;
  const int w = threadIdx.x >> 5;
  const int mt = blockIdx.x;
  const int l15 = lane & 15;
  const int half = lane >> 4;
  const int kofs = half * 2;
  const int row = mt * 16 + l15;
  const int col = w * 16 + l15;
  v8f acc = {};
  for (int seg = 0; seg < nseg; ++seg) {
    const float* S = (seg == 0) ? s0 : ((seg == 1) ? s1 : s2);
    const float* Srow = S + (long)row * FF;
    const float* Wseg = W + (long)seg * FF * FF;
    for (int k = 0; k < FF; k += 4) {
      v2f a = *(const v2f*)(Srow + k + kofs);  // 8B-aligned (kofs even)
      v2f bf;
      bf.x = Wseg[(long)(k + kofs) * FF + col];
      bf.y = Wseg[(long)(k + kofs + 1) * FF + col];
      acc = wmma4(a, bf, acc);
    }
  }
  const float bv = bias[col];
  for (int r = 0; r < 8; ++r) {
    int m = mt * 16 + half * 8 + r;
    float v = acc[r] + bv;
    if (act_tanh) v = tanhf(v);
    out[(long)m * FF + col] = v;
  }
}

// ---------------- adjacency aggregation ----------------
// TRANS=false: out[i,:] = inv[i] * sum_j adj[i,j] * X[j,:]
// TRANS=true : out[i,:] = inv[i] * sum_j adj[j,i] * X[j,:]
// block: 32 output rows x 128 cols; 8 waves, each 16 cols, two 16x16 accs.
// Double-buffered async-to-LDS staging: 5 async b128 copies per thread per
// 32-deep K chunk (1 for the adj tile, 4 for the feature tile).
template <bool TRANS>
__global__ void agg_kernel(const float* __restrict__ adj,
                           const float* __restrict__ X,
                           const float* __restrict__ invnorm,
                           float* __restrict__ out) {
  constexpr int TM = 32;   // output rows per block
  constexpr int KC = 32;   // k chunk
  constexpr int AP = 36;   // lds_a pitch: 32 + 4 pad (distinct banks per lane)
  constexpr int BP = 136;  // lds_b pitch: halves land on disjoint bank groups
  __shared__ float lds_a[2][KC * AP];  // [k-or-m major per TRANS]
  __shared__ float lds_b[2][KC * BP];

  const int b = blockIdx.y;
  const int i0 = blockIdx.x * TM;
  const int tid = threadIdx.x;
  const int lane = tid & 31;
  const int w = tid >> 5;
  const int l15 = lane & 15;
  const int half = lane >> 4;
  const int kofs = half * 2;
  const int n0 = w * 16;

  const float* adjb = adj + (long)b * NN * NN;
  const float* Xb = X + (long)b * NN * FF;

  // per-thread staging coordinates
  const int ar = tid >> 3;         // 0..31: A tile row (m row or k row)
  const int ac = (tid & 7) * 4;    // A tile col (float4)
  const int br = tid >> 5;         // 0..7: B tile row base
  const int bc = (tid & 31) * 4;   // B tile col (float4)

  auto issue = [&](int k0, int db) {
    if (!TRANS) {
      // adj rows i0..i0+31, cols k0..k0+31 -> lds_a[m][k]
      async_cp16(&lds_a[db][ar * AP + ac],
                 &adjb[(long)(i0 + ar) * NN + k0 + ac]);
    } else {
      // adj rows k0..k0+31, cols i0..i0+31 -> lds_a[k][m]
      async_cp16(&lds_a[db][ar * AP + ac],
                 &adjb[(long)(k0 + ar) * NN + i0 + ac]);
    }
    for (int p = 0; p < 4; ++p) {  // X rows k0..k0+31, 128 wide -> lds_b[k][v]
      int rr = p * 8 + br;
      async_cp16(&lds_b[db][rr * BP + bc], &Xb[(long)(k0 + rr) * FF + bc]);
    }
  };

  v8f acc0 = {};
  v8f acc1 = {};

  constexpr int NCH = NN / KC;  // 128 chunks
  issue(0, 0);
  for (int c = 0; c < NCH; ++c) {
    const int db = c & 1;
    if (c + 1 < NCH) {
      issue((c + 1) * KC, db ^ 1);
      wait_async5();  // chunk c's 5 copies retired (in-order completion)
    } else {
      wait_async0();
    }
    __syncthreads();  // every wave's copies for chunk c are in LDS

    const float* A = lds_a[db];
    const float* Bt = lds_b[db];
#pragma unroll
    for (int kk = 0; kk < KC; kk += 4) {
      v2f bf;
      bf.x = Bt[(kk + kofs) * BP + n0 + l15];
      bf.y = Bt[(kk + kofs + 1) * BP + n0 + l15];
      v2f a0, a1;
      if (!TRANS) {
        a0 = *(const v2f*)&A[l15 * AP + kk + kofs];
        a1 = *(const v2f*)&A[(16 + l15) * AP + kk + kofs];
      } else {
        a0.x = A[(kk + kofs) * AP + l15];
        a0.y = A[(kk + kofs + 1) * AP + l15];
        a1.x = A[(kk + kofs) * AP + 16 + l15];
        a1.y = A[(kk + kofs + 1) * AP + 16 + l15];
      }
      acc0 = wmma4(a0, bf, acc0);
      acc1 = wmma4(a1, bf, acc1);
    }
    __syncthreads();  // all reads done before this buffer is refilled
  }

  const float* inv = invnorm + b * NN;
  float* ob = out + (long)b * NN * FF;
  for (int t = 0; t < 2; ++t) {
    v8f acc = t ? acc1 : acc0;
    for (int r = 0; r < 8; ++r) {
      int m = i0 + t * 16 + half * 8 + r;
      ob[(long)m * FF + n0 + l15] = acc[r] * inv[m];
    }
  }
}

extern "C" void kernel_launch(void* const* d_in, const int* in_sizes, int n_in,
                              void* d_out, int out_size, void* d_ws,
                              size_t ws_size, hipStream_t stream) {
  (void)in_sizes; (void)n_in; (void)out_size; (void)ws_size;
  const float* nodes = (const float*)d_in[0];
  const float* adj = (const float*)d_in[1];
  auto P = [&](int l, int j) -> const float* {
    return (const float*)d_in[2 + 8 * l + j];
  };

  // workspace layout (~34 MB of f32)
  float* p = (float*)d_ws;
  float* invr = p;  p += BB * NN;
  float* invc = p;  p += BB * NN;
  float* srcb = p;  p += (long)BB * NN * FF;
  float* dstb = p;  p += (long)BB * NN * FF;
  float* selfb = p; p += (long)BB * NN * FF;
  float* sagg = p;  p += (long)BB * NN * FF;
  float* dagg = p;  p += (long)BB * NN * FF;
  float* outs[4];
  outs[0] = p; p += (long)BB * NN * FF;
  outs[1] = p; p += (long)BB * NN * FF;
  outs[2] = p; p += (long)BB * NN * FF;
  outs[3] = (float*)d_out;

  rownorm_kernel<<<dim3(NN / 8, BB), 256, 0, stream>>>(adj, invr);
  colnorm_kernel<<<dim3(NN / 256, BB), 256, 0, stream>>>(adj, invc);

  const int MT = BB * NN / 16;  // 512 row tiles

  for (int l = 0; l < 4; ++l) {
    const float *x0, *x1, *x2;
    int nseg;
    if (l == 0)      { x0 = nodes;   x1 = x0;      x2 = x0;     nseg = 1; }
    else if (l == 1) { x0 = outs[0]; x1 = x0;      x2 = x0;     nseg = 1; }
    else if (l == 2) { x0 = outs[1]; x1 = outs[0]; x2 = x1;     nseg = 2; }
    else             { x0 = outs[2]; x1 = outs[0]; x2 = nodes;  nseg = 3; }

    gemm_seg_kernel<<<dim3(MT), 256, 0, stream>>>(x0, x1, x2, P(l, 0), P(l, 1),
                                                  srcb, nseg, 0);
    gemm_seg_kernel<<<dim3(MT), 256, 0, stream>>>(x0, x1, x2, P(l, 2), P(l, 3),
                                                  dstb, nseg, 0);
    gemm_seg_kernel<<<dim3(MT), 256, 0, stream>>>(x0, x1, x2, P(l, 4), P(l, 5),
                                                  selfb, nseg, 0);

    agg_kernel<false><<<dim3(NN / 32, BB), 256, 0, stream>>>(adj, srcb, invr,
                                                             sagg);
    agg_kernel<true><<<dim3(NN / 32, BB), 256, 0, stream>>>(adj, dstb, invc,
                                                            dagg);

    gemm_seg_kernel<<<dim3(MT), 256, 0, stream>>>(sagg, selfb, dagg, P(l, 6),
                                                  P(l, 7), outs[l], 3, 1);
  }
}